// PointPillarsModel_41875931136631
// MI455X (gfx1250) — compile-verified
//
#include <hip/hip_runtime.h>
#include <stdint.h>

// Problem shape (fixed by setup_inputs in the reference):
//   features: (BS=4, C=64, P=12000) f32
//   x_orig_indices / y_orig_indices: (BS, P) int64
//   output: (BS, C, NY=440, NX=500) f32  (NX/NY read from device scalars)
#define BS_  4
#define C_   64
#define P_   12000
#define TPB  256
#define CSPLIT 4
#define CPB  (C_ / CSPLIT)   // 16 channels per block

// ---------------------------------------------------------------------------
// Kernel 1: zero-fill the 225 MB output with 128-bit stores (grid-stride).
// ---------------------------------------------------------------------------
__global__ __launch_bounds__(TPB) void pp_zero_kernel(float4* __restrict__ out4,
                                                      int n4,
                                                      float* __restrict__ out,
                                                      int n_total) {
    int i = blockIdx.x * blockDim.x + threadIdx.x;
    const int stride = gridDim.x * blockDim.x;
    const float4 z = make_float4(0.f, 0.f, 0.f, 0.f);
    for (; i < n4; i += stride) out4[i] = z;                 // global_store_b128
    // scalar tail (n_total not divisible by 4 — not the case here, but safe)
    if (blockIdx.x == 0 && threadIdx.x == 0) {
        for (int t = n4 * 4; t < n_total; ++t) out[t] = 0.f;
    }
}

// ---------------------------------------------------------------------------
// Kernel 2: scatter. One thread per point; CPB channels per thread.
// Index pairs staged via CDNA5 async global->LDS (ASYNCcnt path).
// ---------------------------------------------------------------------------
__global__ __launch_bounds__(TPB) void pp_scatter_kernel(
    const float* __restrict__ feat,       // [BS][C][P]
    const long long* __restrict__ xig,    // [BS][P] int64
    const long long* __restrict__ yig,    // [BS][P] int64
    const int* __restrict__ nxp,          // scalar NX (device)
    const int* __restrict__ nyp,          // scalar NY (device)
    float* __restrict__ out)              // [BS][C][NY][NX]
{
    __shared__ long long s_xi[TPB];
    __shared__ long long s_yi[TPB];

    const int tid = threadIdx.x;
    const int b   = blockIdx.y;
    const int p   = blockIdx.x * TPB + tid;
    const int c0  = blockIdx.z * CPB;
    const int NX  = nxp[0];               // low 32 bits; valid for int32 or int64 LE
    const int NY  = nyp[0];

    if (p < P_) {
        // LDS byte address = low 32 bits of the shared->flat pointer
        unsigned lds_x = (unsigned)(uintptr_t)&s_xi[tid];
        unsigned lds_y = (unsigned)(uintptr_t)&s_yi[tid];
        unsigned boff  = (unsigned)((b * P_ + p) * 8);   // int64 byte offset, < 2^31
        // GVS mode: address is the SGPR base plus the per-lane 32-bit VGPR offset
        asm volatile("global_load_async_to_lds_b64 %0, %1, %2"
                     :: "v"(lds_x), "v"(boff), "s"(xig) : "memory");
        asm volatile("global_load_async_to_lds_b64 %0, %1, %2"
                     :: "v"(lds_y), "v"(boff), "s"(yig) : "memory");
    }
    asm volatile("s_wait_asynccnt 0x0" ::: "memory");
    __syncthreads();

    if (p >= P_) return;

    long long xi = s_xi[tid];
    long long yi = s_yi[tid];
    xi = xi < 0 ? 0 : (xi > (long long)(NX - 1) ? (long long)(NX - 1) : xi);
    yi = yi < 0 ? 0 : (yi > (long long)(NY - 1) ? (long long)(NY - 1) : yi);

    const size_t plane = (size_t)NX * (size_t)NY;
    const size_t off   = (size_t)yi * (size_t)NX + (size_t)xi;

    const float* f = feat + ((size_t)b * C_ + (size_t)c0) * (size_t)P_ + (size_t)p;
    float*       o = out  + ((size_t)b * C_ + (size_t)c0) * plane + off;

    // Coalesced feature reads (lanes = consecutive points); random 4B stores.
    #pragma unroll
    for (int c = 0; c < CPB; ++c) {
        o[(size_t)c * plane] = f[(size_t)c * P_];
    }
}

// ---------------------------------------------------------------------------
extern "C" void kernel_launch(void* const* d_in, const int* in_sizes, int n_in,
                              void* d_out, int out_size, void* d_ws, size_t ws_size,
                              hipStream_t stream) {
    const float*     feat = (const float*)d_in[0];
    const long long* xig  = (const long long*)d_in[1];
    const long long* yig  = (const long long*)d_in[2];
    const int*       nxp  = (const int*)d_in[3];
    const int*       nyp  = (const int*)d_in[4];
    float*           out  = (float*)d_out;

    // Zero-fill: 14.08M float4 stores, grid-stride over up to 4096 blocks.
    int n4 = out_size / 4;
    int zb = (n4 + TPB - 1) / TPB;
    if (zb > 4096) zb = 4096;
    if (zb < 1) zb = 1;
    pp_zero_kernel<<<zb, TPB, 0, stream>>>((float4*)out, n4, out, out_size);

    // Scatter: 47 point-blocks x 4 batches x 4 channel-slices = 752 blocks.
    dim3 grid((P_ + TPB - 1) / TPB, BS_, CSPLIT);
    pp_scatter_kernel<<<grid, TPB, 0, stream>>>(feat, xig, yig, nxp, nyp, out);
}